// Model_50869592654042
// MI455X (gfx1250) — compile-verified
//
#include <hip/hip_runtime.h>
#include <math.h>

typedef __attribute__((ext_vector_type(2))) float v2f;
typedef __attribute__((ext_vector_type(8))) float v8f;

#define NTRI       1000
#define TBLK       250            // blocks of 4 triangles -> one 16x16x4 WMMA each
#define COEF_ELEMS (TBLK * 32)    // 8000 float2 entries, A-operand lane layout
#define H          256
// orient is multiplied in per-edge; this is 1/SHARP * log2(e) so that
// sigmoid(d/SHARP) = 1/(1+exp2(-d*SCALE)) and v_exp_f32 needs no pre-multiply.
#define SHARP_LOG2 144.26950408889634f
#define VIEW_W     0.57735026918962576f   // tan(30 deg)

// ---------------------------------------------------------------------------
// Kernel A: per (triangle-block, lane) precompute WMMA A-operand coefficients.
// lane 0-15 : (A*orient*SHARP_LOG2, B*orient*SHARP_LOG2) row M=lane    (K=0,1)
// lane16-31 : (C*orient*SHARP_LOG2, 0)                   row M=lane-16 (K=2,3)
// Row map inside block t: M 0-2 -> tri 4t, M 3-5 -> 4t+1, M 6,7 pad,
//                          M 8-10 -> 4t+2, M 11-13 -> 4t+3, M 14,15 pad.
// ---------------------------------------------------------------------------
__global__ void edge_setup(const float* __restrict__ verts,
                           const int*   __restrict__ faces,
                           const float* __restrict__ cam,
                           float2*      __restrict__ coef)
{
    int tid = blockIdx.x * blockDim.x + threadIdx.x;
    if (tid >= COEF_ELEMS) return;
    int t    = tid >> 5;
    int lane = tid & 31;
    int M    = lane & 15;
    bool isC = (lane >= 16);
    int grp  = M >> 3;
    int sub  = M & 7;

    float2 out; out.x = 0.0f; out.y = 0.0f;
    if (sub >= 6) { coef[tid] = out; return; }          // pad rows: never read
    int tri  = t * 4 + grp * 2 + sub / 3;
    int edge = sub % 3;
    if (tri >= NTRI) { coef[tid] = out; return; }

    // camera basis (cheap redundant per-thread compute)
    float exm = cam[0], eym = cam[1], ezm = cam[2];
    float nz  = sqrtf(exm*exm + eym*eym + ezm*ezm) + 1e-8f;
    float zx = -exm/nz, zy = -eym/nz, zz = -ezm/nz;
    float cx = zz, czv = -zx;                            // cross([0,1,0], z) = (zz, 0, -zx)
    float nx = sqrtf(cx*cx + czv*czv) + 1e-8f;
    float xxv = cx/nx, xyv = 0.0f, xzv = czv/nx;
    float yxv = zy*xzv - zz*xyv;
    float yyv = zz*xxv - zx*xzv;
    float yzv = zx*xyv - zy*xxv;

    int i0 = faces[tri*3+0], i1 = faces[tri*3+1], i2 = faces[tri*3+2];
    int idxs[3] = { i0, i1, i2 };
    float X[3], Y[3];
#pragma unroll
    for (int k = 0; k < 3; ++k) {
        float px = verts[idxs[k]*3+0] - exm;
        float py = verts[idxs[k]*3+1] - eym;
        float pz = verts[idxs[k]*3+2] - ezm;
        float vcx = xxv*px + xyv*py + xzv*pz;
        float vcy = yxv*px + yyv*py + yzv*pz;
        float vcz = zx*px  + zy*py  + zz*pz;
        float den = vcz * VIEW_W + 1e-8f;
        X[k] = vcx / den;
        Y[k] = vcy / den;
    }
    int e0 = edge, e1 = (edge + 1) % 3;
    float v0x = X[e0], v0y = Y[e0];
    float ex_ = X[e1] - v0x, ey_ = Y[e1] - v0y;
    float elen = sqrtf(ex_*ex_ + ey_*ey_) + 1e-8f;
    float Aq =  ex_ / elen;
    float Bq = -ey_ / elen;
    float Cq = (ey_*v0x - ex_*v0y) / elen;
    float eax = X[1]-X[0], eay = Y[1]-Y[0];
    float ebx = X[2]-X[0], eby = Y[2]-Y[0];
    float det = eax*eby - eay*ebx;
    float orient = (det > 0.f) ? 1.f : ((det < 0.f) ? -1.f : 0.f);
    float s = orient * SHARP_LOG2;
    if (isC) { out.x = Cq * s; out.y = 0.0f;   }
    else     { out.x = Aq * s; out.y = Bq * s; }
    coef[tid] = out;
}

// ---------------------------------------------------------------------------
// Kernel B: main pixel x triangle loop. One wave = 16 pixels x all triangles.
// D = A(16 edges x K4) x B(K4 x 16 pixels) via v_wmma_f32_16x16x4_f32, then
// pure-TRANS sigmoid/log work in the log2 domain (no constant multiplies).
// A-operand LDS loads are prefetched 2 blocks ahead to hide ds latency.
// ---------------------------------------------------------------------------
__global__ void sil_loss(const float*  __restrict__ imref,
                         const float2* __restrict__ coef,
                         float*        __restrict__ partials)
{
    __shared__ v2f   sCoef[COEF_ELEMS];   // 64 000 B
    __shared__ float sWave[8];
    const int tid = threadIdx.x;

    for (int i = tid; i < COEF_ELEMS; i += 256) {
        float2 c2 = coef[i];
        v2f v; v.x = c2.x; v.y = c2.y;
        sCoef[i] = v;
    }
    __syncthreads();

    const int lane = tid & 31;
    const int wave = tid >> 5;
    const int gw   = blockIdx.x * 8 + wave;        // global wave id
    const int pix  = gw * 16 + (lane & 15);        // one pixel per lane column
    const int prow = pix >> 8;
    const int pcol = pix & 255;
    const float cy = ((float)prow + 0.5f) * (2.0f / H) - 1.0f;   // py (multiplies A)
    const float cx = ((float)pcol + 0.5f) * (2.0f / H) - 1.0f;   // px (multiplies B)
    const bool hi  = (lane >= 16);

    // B operand: K rows (py, px, 1, 0) striped: VGPR0 = K0/K2, VGPR1 = K1/K3
    v2f bmat;
    bmat.x = hi ? 1.0f : cy;
    bmat.y = hi ? 0.0f : cx;

    float lb2 = 0.0f;                      // sum of log2(1 - prob)
    // distance-2 rotating prefetch of the A operand
    v2f a0 = sCoef[lane];
    v2f a1 = sCoef[32 + lane];
#pragma unroll 2
    for (int t = 0; t < TBLK; ++t) {
        v2f a = a0;
        a0 = a1;
        int tn = (t + 2 < TBLK) ? (t + 2) : (TBLK - 1);
        a1 = sCoef[tn * 32 + lane];        // issued ~1 full iteration early

        v8f cacc = {0.f,0.f,0.f,0.f,0.f,0.f,0.f,0.f};
        v8f d = __builtin_amdgcn_wmma_f32_16x16x4_f32(
            /*neg_a=*/false, a, /*neg_b=*/false, bmat,
            /*c_mod=*/(short)0, cacc, /*reuse_a=*/false, /*reuse_b=*/false);

        // rows 0-2 / 3-5 of this lane's half = 2 triangles; rows 6,7 are pad.
        // e_i = exp2(-d_i) == exp(-d_orig/SHARP); negation is a src modifier.
        float e0 = __builtin_amdgcn_exp2f(-d[0]);
        float e1 = __builtin_amdgcn_exp2f(-d[1]);
        float e2 = __builtin_amdgcn_exp2f(-d[2]);
        float e3 = __builtin_amdgcn_exp2f(-d[3]);
        float e4 = __builtin_amdgcn_exp2f(-d[4]);
        float e5 = __builtin_amdgcn_exp2f(-d[5]);
        float q0 = (1.0f + e0) * (1.0f + e1) * (1.0f + e2);
        float q1 = (1.0f + e3) * (1.0f + e4) * (1.0f + e5);
        float p0 = fminf(__builtin_amdgcn_rcpf(q0), 0.999999f);
        float p1 = fminf(__builtin_amdgcn_rcpf(q1), 0.999999f);
        lb2 += __builtin_amdgcn_logf(1.0f - p0);
        lb2 += __builtin_amdgcn_logf(1.0f - p1);
    }

    // combine the two half-wave triangle partitions for this pixel
    float lbTot = lb2 + __shfl_xor(lb2, 16, 32);

    float contrib = 0.0f;
    if (!hi) {
        // log_bg (natural) = lbTot * ln2, so exp(log_bg) = exp2(lbTot)
        float sil = 1.0f - __builtin_amdgcn_exp2f(lbTot);
        float dd  = sil - imref[pix];
        contrib = dd * dd;
    }
    for (int off = 16; off > 0; off >>= 1)
        contrib += __shfl_down(contrib, off, 32);
    if (lane == 0) sWave[wave] = contrib;
    __syncthreads();
    if (tid == 0) {
        float s = 0.0f;
#pragma unroll
        for (int wv = 0; wv < 8; ++wv) s += sWave[wv];
        partials[blockIdx.x] = s;
    }
}

// ---------------------------------------------------------------------------
// Kernel C: deterministic fixed-tree final reduction of 512 block partials.
// ---------------------------------------------------------------------------
__global__ void reduce_final(const float* __restrict__ partials,
                             float*       __restrict__ out)
{
    __shared__ float sd[256];
    int tid = threadIdx.x;
    sd[tid] = partials[tid] + partials[tid + 256];
    __syncthreads();
    for (int s = 128; s > 0; s >>= 1) {
        if (tid < s) sd[tid] += sd[tid + s];
        __syncthreads();
    }
    if (tid == 0) out[0] = sd[0];
}

extern "C" void kernel_launch(void* const* d_in, const int* in_sizes, int n_in,
                              void* d_out, int out_size, void* d_ws, size_t ws_size,
                              hipStream_t stream)
{
    const float* verts = (const float*)d_in[0];   // (1,5000,3) f32
    const int*   faces = (const int*)  d_in[1];   // (1,1000,3) i32
    const float* cam   = (const float*)d_in[2];   // (3,)       f32
    const float* imref = (const float*)d_in[3];   // (256,256)  f32

    float2* coef     = (float2*)d_ws;                                       // 64 000 B
    float*  partials = (float*)((char*)d_ws + COEF_ELEMS * sizeof(float2)); // 2 048 B

    edge_setup<<<(COEF_ELEMS + 127) / 128, 128, 0, stream>>>(verts, faces, cam, coef);
    sil_loss<<<512, 256, 0, stream>>>(imref, coef, partials);
    reduce_final<<<1, 256, 0, stream>>>(partials, (float*)d_out);
}